// ResamplingModel_55860344651821
// MI455X (gfx1250) — compile-verified
//
#include <hip/hip_runtime.h>
#include <hip/hip_bf16.h>
#include <stdint.h>

#define B_SZ   256
#define N_SZ   16384
#define TPB    1024          // 32 waves of 32
#define EPT    16            // elements per thread
#define ALPHA_F 0.5f

typedef __attribute__((ext_vector_type(2))) float v2f;
typedef __attribute__((ext_vector_type(8))) float v8f;

#define AS1 __attribute__((address_space(1)))
#define AS3 __attribute__((address_space(3)))

#if defined(__has_builtin)
#if __has_builtin(__builtin_amdgcn_global_load_async_to_lds_b32) && \
    __has_builtin(__builtin_amdgcn_s_wait_asynccnt)
#define HAVE_ASYNC_LDS 1
#endif
#endif

// ---------------------------------------------------------------- threefry
__device__ __forceinline__ uint32_t rotl32(uint32_t x, int r) {
    return (x << r) | (x >> (32 - r));
}

// Exact JAX threefry-2x32 with key(42) -> (0,42); flat counter g in [0, B*N).
// JAX pairs counter halves: (q, q + 2^21); output half selected by g >= 2^21.
__device__ __forceinline__ float jax_uniform_u01(uint32_t g) {
    const uint32_t HALF = 2097152u;           // (B*N)/2 = 2^21
    uint32_t q  = g & (HALF - 1u);
    uint32_t x0 = q, x1 = q + HALF;
    const uint32_t ks0 = 0u, ks1 = 42u, ks2 = 0x1BD11BDAu ^ 0u ^ 42u;
    x0 += ks0; x1 += ks1;
#define TF_R(r) { x0 += x1; x1 = rotl32(x1, r); x1 ^= x0; }
    TF_R(13) TF_R(15) TF_R(26) TF_R(6)   x0 += ks1; x1 += ks2 + 1u;
    TF_R(17) TF_R(29) TF_R(16) TF_R(24)  x0 += ks2; x1 += ks0 + 2u;
    TF_R(13) TF_R(15) TF_R(26) TF_R(6)   x0 += ks0; x1 += ks1 + 3u;
    TF_R(17) TF_R(29) TF_R(16) TF_R(24)  x0 += ks1; x1 += ks2 + 4u;
    TF_R(13) TF_R(15) TF_R(26) TF_R(6)   x0 += ks2; x1 += ks0 + 5u;
#undef TF_R
    uint32_t bits = (g < HALF) ? x0 : x1;
    return __uint_as_float((bits >> 9) | 0x3F800000u) - 1.0f;
}

// ------------------------------------------------------- WMMA 256-elem scan
// Inclusive scan of sh[base .. base+255], viewed column-major as X[m][n] =
// sh[base + n*16 + m].  Y = L_incl*X  +  bcast(colTotals)*U_strict, each
// 16x16x16 product chained as 4x V_WMMA_F32_16X16X4_F32 (K=4).
// A operand layout (16x4 f32): lane L, vgpr v -> M = L%16, K = 2*(L/16)+v.
// B operand layout (4x16 f32): lane L, vgpr v -> N = L%16, K = 2*(L/16)+v.
// C/D layout (16x16 f32):      lane L, vgpr v -> M = v + 8*(L/16), N = L%16.
__device__ __forceinline__ float wmma_scan256(float* sh, int base, float carry,
                                              int lane) {
    const int lo = lane & 15;
    const int hi = lane >> 4;
    v8f acc = {};
    // P1 = L_incl (A, constant) x X (B, from LDS): within-column inclusive scan
#pragma unroll
    for (int kc = 0; kc < 4; ++kc) {
        const int k0 = 4 * kc + 2 * hi;
        v2f a, bm;
        a[0]  = (k0     <= lo) ? 1.0f : 0.0f;   // L_incl[m][k] = (k <= m)
        a[1]  = (k0 + 1 <= lo) ? 1.0f : 0.0f;
        bm[0] = sh[base + lo * 16 + k0];        // X[k][n], n = lo
        bm[1] = sh[base + lo * 16 + k0 + 1];
        acc = __builtin_amdgcn_wmma_f32_16x16x4_f32(false, a, false, bm,
                                                    (short)0, acc, false, false);
    }
    // column totals ct[c] = P1[15][c] live in lane 16+c, vgpr 7
    const float p15 = acc[7];
    // acc += bcast(ct) (A) x U_strict (B, constant): cross-column offsets
#pragma unroll
    for (int kc = 0; kc < 4; ++kc) {
        const int k0 = 4 * kc + 2 * hi;
        v2f a, bm;
        a[0]  = __shfl(p15, 16 + k0, 32);
        a[1]  = __shfl(p15, 16 + k0 + 1, 32);
        bm[0] = (k0     < lo) ? 1.0f : 0.0f;    // U_strict[k][n] = (k < n)
        bm[1] = (k0 + 1 < lo) ? 1.0f : 0.0f;
        acc = __builtin_amdgcn_wmma_f32_16x16x4_f32(false, a, false, bm,
                                                    (short)0, acc, false, false);
    }
#pragma unroll
    for (int v = 0; v < 8; ++v)
        sh[base + lo * 16 + 8 * hi + v] = acc[v] + carry;
    // chunk running total = element 255 = lane 31, vgpr 7 (carry included)
    return __shfl(acc[7] + carry, 31, 32);
}

// -------------------------------------------------------- block reductions
__device__ __forceinline__ float block_reduce(float v, int ismax, float* scr) {
    const int lane = threadIdx.x & 31, wv = threadIdx.x >> 5;
#pragma unroll
    for (int o = 16; o; o >>= 1) {
        float x = __shfl_xor(v, o, 32);
        v = ismax ? fmaxf(v, x) : (v + x);
    }
    if (lane == 0) scr[wv] = v;
    __syncthreads();
    if (wv == 0) {
        float u = scr[lane];
#pragma unroll
        for (int o = 16; o; o >>= 1) {
            float x = __shfl_xor(u, o, 32);
            u = ismax ? fmaxf(u, x) : (u + x);
        }
        if (lane == 0) scr[0] = u;
    }
    __syncthreads();
    float r = scr[0];
    __syncthreads();              // scr reusable after return
    return r;
}

// ------------------------------------------------------------------ kernel
__global__ __launch_bounds__(TPB) void ResamplingModel_55860344651821_kernel(
    const float* __restrict__ states, const float* __restrict__ weights,
    float* __restrict__ out_states, float* __restrict__ out_weights) {
    __shared__ float sh[N_SZ];      // weights row -> exp(q) -> CDF
    __shared__ int   shCnt[N_SZ];   // multinomial counts
    __shared__ float shScr[32];     // reduce scratch / wave totals
    __shared__ float shStat[2];     // logZ, logS2

    const int b    = blockIdx.x;
    const int t    = threadIdx.x;
    const int lane = t & 31;
    const int wv   = t >> 5;
    const int base = t * EPT;
    const float* wrow = weights + (size_t)b * N_SZ;

    // ---- stage 0: weights row -> LDS (async TDM-style path if available)
#if HAVE_ASYNC_LDS
#pragma unroll
    for (int j = 0; j < EPT; ++j) {
        const int idx = t + j * TPB;               // coalesced per issue
        __builtin_amdgcn_global_load_async_to_lds_b32(
            (AS1 int*)(int*)(wrow + idx), (AS3 int*)(int*)(sh + idx), 0, 0);
    }
    __builtin_amdgcn_s_wait_asynccnt(0);
#else
#pragma unroll
    for (int j = 0; j < EPT; ++j) {
        const int idx = t + j * TPB;
        sh[idx] = wrow[idx];
    }
#endif
    __syncthreads();

    // ---- stage 1: logsumexp of weights; q = log(alpha*e^pw + (1-a)/n)
    float wvreg[EPT];
#pragma unroll
    for (int j = 0; j < EPT; ++j) wvreg[j] = sh[base + j];

    float m = -__builtin_inff();
#pragma unroll
    for (int j = 0; j < EPT; ++j) m = fmaxf(m, wvreg[j]);
    m = block_reduce(m, 1, shScr);

    float s = 0.0f;
#pragma unroll
    for (int j = 0; j < EPT; ++j) s += __expf(wvreg[j] - m);
    s = block_reduce(s, 0, shScr);
    const float logZ = m + __logf(s);

    const float c1 = (1.0f - ALPHA_F) / (float)N_SZ;
    float eq[EPT];
    float s2p = 0.0f;
#pragma unroll
    for (int j = 0; j < EPT; ++j) {
        const float pw = wvreg[j] - logZ;
        eq[j] = ALPHA_F * __expf(pw) + c1;   // exp(q) before normalization
        s2p  += eq[j];
    }
    const float S2    = block_reduce(s2p, 0, shScr);
    const float logS2 = __logf(S2);
    if (t == 0) { shStat[0] = logZ; shStat[1] = logS2; }

#pragma unroll
    for (int j = 0; j < EPT; ++j) sh[base + j] = eq[j];
    __syncthreads();

    // ---- stage 2: 16384-wide prefix sum via WMMA tiles + cross-wave offsets
    float carry = 0.0f;
#pragma unroll
    for (int c = 0; c < 2; ++c)
        carry = wmma_scan256(sh, wv * 512 + c * 256, carry, lane);
    if (lane == 0) shScr[wv] = carry;      // wave totals
    __syncthreads();
    float woff = 0.0f;
    for (int j = 0; j < wv; ++j) woff += shScr[j];
#pragma unroll
    for (int j = 0; j < EPT; ++j) sh[base + j] += woff;   // full (unnorm.) CDF
#pragma unroll
    for (int j = 0; j < EPT; ++j) shCnt[base + j] = 0;
    __syncthreads();

    // ---- stage 3: n inverse-CDF draws (exact JAX threefry), bincount in LDS
#pragma unroll 4
    for (int j = 0; j < EPT; ++j) {
        const uint32_t g = (uint32_t)(b * N_SZ + base + j);
        const float u = jax_uniform_u01(g) * S2;   // search unnormalized CDF
        int lo2 = 0, hi2 = N_SZ;
        while (lo2 < hi2) {                         // searchsorted, side=left
            const int mid = (lo2 + hi2) >> 1;
            if (sh[mid] < u) lo2 = mid + 1; else hi2 = mid;
        }
        const int d = (lo2 < N_SZ) ? lo2 : (N_SZ - 1);
        atomicAdd(&shCnt[d], 1);
    }
    __syncthreads();

    // ---- stage 4: faithful gather with COUNTS used as indices (clamped)
    const float logZb  = shStat[0];
    const float logS2b = shStat[1];
    const size_t rowoff = (size_t)b * N_SZ;
#pragma unroll
    for (int j = 0; j < EPT; ++j) {
        const int i = base + j;
        long long fidx = (long long)shCnt[i] + (long long)rowoff;
        const long long maxi = (long long)B_SZ * N_SZ - 1;
        if (fidx > maxi) fidx = maxi;
        const size_t src = (size_t)fidx;
        const size_t dst = rowoff + (size_t)i;
        out_states[dst * 3 + 0] = states[src * 3 + 0];
        out_states[dst * 3 + 1] = states[src * 3 + 1];
        out_states[dst * 3 + 2] = states[src * 3 + 2];
        // new_weights[src] = pw - q = pw - log(a*e^pw + c1) + logS2
        const float ws = weights[src];
        const float pw = ws - logZb;
        out_weights[dst] = pw - __logf(ALPHA_F * __expf(pw) + c1) + logS2b;
    }
}

extern "C" void kernel_launch(void* const* d_in, const int* in_sizes, int n_in,
                              void* d_out, int out_size, void* d_ws,
                              size_t ws_size, hipStream_t stream) {
    (void)in_sizes; (void)n_in; (void)d_ws; (void)ws_size; (void)out_size;
    const float* states  = (const float*)d_in[0];
    const float* weights = (const float*)d_in[1];
    float* out_states  = (float*)d_out;
    float* out_weights = (float*)d_out + (size_t)B_SZ * N_SZ * 3;
    ResamplingModel_55860344651821_kernel<<<B_SZ, TPB, 0, stream>>>(
        states, weights, out_states, out_weights);
}